// AugmentedTripletLoss_40157944218131
// MI455X (gfx1250) — compile-verified
//
#include <hip/hip_runtime.h>
#include <math.h>

#define ALPHA 0.1f
#define BETA  1.1f
#define EPSN  1e-8f
#define NCLS  16
#define DIM   512
#define NSAMP 131072
#define KCH   64     // K-chunk staged through LDS per wave

typedef float v2f __attribute__((ext_vector_type(2)));
typedef float v4f __attribute__((ext_vector_type(4)));
typedef float v8f __attribute__((ext_vector_type(8)));

// workspace layout (float offsets)
#define OFF_SUMS  0          // 16*512
#define OFF_CNT   8192       // 16
#define OFF_CENTN 8224       // 16*512 normalized centroids
#define OFF_DEG   16416      // 16
#define OFF_PM    16448      // 16*16
#define OFF_S     16704      // 16*16
#define OFF_INTRA 16960      // 1
#define WS_FLOATS 16961

__global__ void k_zero(float* __restrict__ ws) {
  int i = blockIdx.x * blockDim.x + threadIdx.x;
  if (i < WS_FLOATS) ws[i] = 0.0f;
}

// ---- Pass 1: per-class sums + counts (full 256MB read, lane-coalesced) ----
__global__ void __launch_bounds__(256) k_centroid_sums(
    const float* __restrict__ emb, const int* __restrict__ labels,
    float* __restrict__ ws) {
  __shared__ float lsum[NCLS * DIM];   // 32 KB
  __shared__ float lcnt[NCLS];
  int tid = threadIdx.x;
  for (int i = tid; i < NCLS * DIM; i += 256) lsum[i] = 0.0f;
  if (tid < NCLS) lcnt[tid] = 0.0f;
  __syncthreads();

  int per = NSAMP / gridDim.x;
  int s0  = blockIdx.x * per;
  for (int s = s0; s < s0 + per; ++s) {
    int lab = labels[s];
    const float* row = emb + (size_t)s * DIM;
    // thread t exclusively owns LDS columns {t, t+256} of every class row:
    // no atomics needed, coalesced 128B-per-wave global reads.
    lsum[lab * DIM + tid]       += row[tid];
    lsum[lab * DIM + tid + 256] += row[tid + 256];
    if (tid == 0) lcnt[lab] += 1.0f;
  }
  __syncthreads();

  float* gs = ws + OFF_SUMS;
  for (int i = tid; i < NCLS * DIM; i += 256) atomicAdd(&gs[i], lsum[i]);
  if (tid < NCLS) atomicAdd(&ws[OFF_CNT + tid], lcnt[tid]);
}

// ---- Tiny: centroids -> normalized centroids, pair mask, deg (1 block) ----
__global__ void __launch_bounds__(256) k_pairs(float* __restrict__ ws) {
  int tid = threadIdx.x;
  const float* sums = ws + OFF_SUMS;
  const float* cnt  = ws + OFF_CNT;
  float* centn = ws + OFF_CENTN;
  float* deg   = ws + OFF_DEG;
  float* pm    = ws + OFF_PM;

  if (tid < NCLS) {
    float invc = 1.0f / fmaxf(cnt[tid], 1.0f);
    float sq = 0.0f;
    for (int d = 0; d < DIM; ++d) {
      float v = sums[tid * DIM + d] * invc;
      centn[tid * DIM + d] = v;
      sq = fmaf(v, v, sq);
    }
    float invn = 1.0f / fmaxf(sqrtf(sq), EPSN);
    for (int d = 0; d < DIM; ++d) centn[tid * DIM + d] *= invn;
  }
  __syncthreads();
  {
    int i = tid >> 4, j = tid & 15;
    float dot = 0.0f;
    for (int d = 0; d < DIM; ++d)
      dot = fmaf(centn[i * DIM + d], centn[j * DIM + d], dot);
    float pd = 1.0f - dot;
    bool ok = (i < j) && (pd <= BETA) && (cnt[i] > 0.0f) && (cnt[j] > 0.0f);
    pm[tid] = ok ? 1.0f : 0.0f;
  }
  __syncthreads();
  if (tid < NCLS) {
    float s = 0.0f;
    for (int j = 0; j < NCLS; ++j) s += pm[tid * NCLS + j] + pm[j * NCLS + tid];
    deg[tid] = s;
  }
}

// ---- Pass 2: WMMA distance pass. B staged through LDS with coalesced
//      b128 loads (4 full cachelines / instruction) instead of 16B-granule
//      strided b64s, then fed to WMMA via ds_load in the ISA lane layout. ----
__global__ void __launch_bounds__(256) k_main(
    const float* __restrict__ emb, const int* __restrict__ labels,
    float* __restrict__ ws) {
  __shared__ float lcent[NCLS * DIM];      // 32 KB normalized centroids
  __shared__ float lB[8][NCLS * KCH];      // 8 waves x 4 KB wave-private staging
  __shared__ float lS[NCLS * NCLS];
  __shared__ float lintra;
  int tid = threadIdx.x;
  const float* centn = ws + OFF_CENTN;
  for (int i = tid; i < NCLS * DIM; i += 256) lcent[i] = centn[i];
  for (int i = tid; i < NCLS * NCLS; i += 256) lS[i] = 0.0f;
  if (tid == 0) lintra = 0.0f;
  __syncthreads();

  const float* gdeg = ws + OFF_DEG;
  const int lane   = tid & 31;
  const int wslot  = tid >> 5;
  const int wid    = blockIdx.x * (256 / 32) + wslot;
  const int nwav   = gridDim.x * (256 / 32);
  const int nr     = lane & 15;            // sample slot (N) and class row (M)
  const int koff   = (lane < 16) ? 0 : 2;  // K-parity half per ISA A/B layout
  const int mbase  = (lane < 16) ? 0 : 8;  // C/D layout: VGPR i -> M = i + mbase
  const int srow   = lane >> 4;            // staging: which of the 2 rows/instr
  const int scol   = (lane & 15) * 4;      // staging: 16B column window
  float* myB = lB[wslot];

  float intra_acc = 0.0f;
  const int NTILES = NSAMP / 16;
  for (int tile = wid; tile < NTILES; tile += nwav) {
    const int n0 = tile * 16;
    v8f acc = {0.f, 0.f, 0.f, 0.f, 0.f, 0.f, 0.f, 0.f};
    float sq = 0.0f;
#pragma unroll 1
    for (int kc = 0; kc < DIM; kc += KCH) {
      // stage 16 rows x 64 K: each b128 covers two rows, 256B contiguous each
#pragma unroll
      for (int ri = 0; ri < 8; ++ri) {
        int r = 2 * ri + srow;
        v4f v = *(const v4f*)(emb + (size_t)(n0 + r) * DIM + kc + scol);
        *(v4f*)(&myB[r * KCH + scol]) = v;
      }
      // 16 WMMA steps over this chunk (per-wave LDS ops are in-order)
#pragma unroll
      for (int kb = 0; kb < KCH; kb += 4) {
        v2f b; b.x = myB[nr * KCH + kb + koff];
               b.y = myB[nr * KCH + kb + koff + 1];
        v2f a; a.x = lcent[nr * DIM + kc + kb + koff];
               a.y = lcent[nr * DIM + kc + kb + koff + 1];
        sq = fmaf(b.x, b.x, sq);
        sq = fmaf(b.y, b.y, sq);
        // D = A(16x4 centroids) * B(4x16 samples) + C, f32
        acc = __builtin_amdgcn_wmma_f32_16x16x4_f32(
            false, a, false, b, (short)0, acc, false, false);
      }
    }
    // full ||x||^2: this lane holds K%4 in {koff,koff+1}; lane^16 has the rest
    sq += __shfl_xor(sq, 16, 32);
    float invn = 1.0f / fmaxf(sqrtf(sq), EPSN);
    int   lab  = labels[n0 + nr];
    float degv = gdeg[lab];
#pragma unroll
    for (int vg = 0; vg < 8; ++vg) {
      int   m = mbase + vg;                  // class index of this accumulator
      float d = 1.0f - acc[vg] * invn;       // cosine distance D[m][n]
      if (m == lab) intra_acc += degv * fmaxf(d - ALPHA, 0.0f);
      atomicAdd(&lS[m * NCLS + lab], fmaxf(BETA - d, 0.0f));
    }
  }
  atomicAdd(&lintra, intra_acc);
  __syncthreads();

  float* gS = ws + OFF_S;
  for (int i = tid; i < NCLS * NCLS; i += 256) atomicAdd(&gS[i], lS[i]);
  if (tid == 0) atomicAdd(&ws[OFF_INTRA], lintra);
}

// ---- Final scalar combine (1 block) ----
__global__ void __launch_bounds__(256) k_final(const float* __restrict__ ws,
                                               float* __restrict__ out) {
  __shared__ float red[256];
  int tid = threadIdx.x;
  const float* pm  = ws + OFF_PM;
  const float* S   = ws + OFF_S;
  const float* deg = ws + OFF_DEG;
  const float* cnt = ws + OFF_CNT;
  int i = tid >> 4, j = tid & 15;
  float p     = pm[tid];
  float inter = p * (S[i * 16 + j] + S[j * 16 + i]);
  float ct    = (tid < NCLS) ? deg[tid] * cnt[tid] : 0.0f;

  red[tid] = inter; __syncthreads();
  for (int s = 128; s > 0; s >>= 1) { if (tid < s) red[tid] += red[tid + s]; __syncthreads(); }
  float inter_sum = red[0]; __syncthreads();

  red[tid] = p; __syncthreads();
  for (int s = 128; s > 0; s >>= 1) { if (tid < s) red[tid] += red[tid + s]; __syncthreads(); }
  float npairs = red[0]; __syncthreads();

  red[tid] = ct; __syncthreads();
  for (int s = 128; s > 0; s >>= 1) { if (tid < s) red[tid] += red[tid + s]; __syncthreads(); }
  if (tid == 0) {
    float denom = fmaxf(red[0], 1.0f);
    float loss  = ws[OFF_INTRA] / denom + inter_sum / denom;
    out[0] = (npairs > 0.0f) ? loss : 0.0f;
  }
}

extern "C" void kernel_launch(void* const* d_in, const int* in_sizes, int n_in,
                              void* d_out, int out_size, void* d_ws, size_t ws_size,
                              hipStream_t stream) {
  const float* emb    = (const float*)d_in[0];
  const int*   labels = (const int*)d_in[1];
  float* out = (float*)d_out;
  float* ws  = (float*)d_ws;
  (void)in_sizes; (void)n_in; (void)out_size; (void)ws_size;

  k_zero<<<(WS_FLOATS + 255) / 256, 256, 0, stream>>>(ws);
  k_centroid_sums<<<512, 256, 0, stream>>>(emb, labels, ws);
  k_pairs<<<1, 256, 0, stream>>>(ws);
  // 1024 blocks * 8 waves = 8192 waves = exactly one 16-sample tile per wave
  k_main<<<1024, 256, 0, stream>>>(emb, labels, ws);
  k_final<<<1, 256, 0, stream>>>(ws, out);
}